// LinearAttend_6734508720467
// MI455X (gfx1250) — compile-verified
//
#include <hip/hip_runtime.h>

// Linear attention, fp32, memory-bound (≈256-320MB traffic @ 23.3TB/s → ~12-14µs floor).
// GEMMs use V_WMMA_F32_16X16X4_F32 (wave32, A=16x4 f32 = 2 VGPRs, B=4x16, C/D=16x16 f32 = 8 VGPRs).
// Single-use streams (v, q, out) use non-temporal hints so k (64MB, read twice) stays in the 192MB L2.

typedef float v2f __attribute__((ext_vector_type(2)));
typedef float v4f __attribute__((ext_vector_type(4)));
typedef float v8f __attribute__((ext_vector_type(8)));

#define NHEADS 32     // b*n = 4*8
#define D      64
#define S      8192
#define SPLITS 8
#define SPAN   (S / SPLITS)   // 1024
#define SCHUNK 64
#define LDSTR  68             // padded LDS row stride (floats): conflict-free WMMA frag reads

// ---------------------------------------------------------------------------
// Kernel 1: per-row stats for k softmax over s: (rowmax, 1/sum(exp)) per (head,d)
// ---------------------------------------------------------------------------
__global__ void kstats_kernel(const float* __restrict__ k, float* __restrict__ stats) {
  const int row = blockIdx.x;                 // 0..2047
  const float* kr = k + (size_t)row * S;
  const int t = threadIdx.x;                  // 256 threads
  v4f vals[8];
  float m = -3.4e38f;
#pragma unroll
  for (int i = 0; i < 8; ++i) {
    v4f x = *(const v4f*)&kr[i * 1024 + t * 4];   // b128, coalesced
    vals[i] = x;
    m = fmaxf(m, fmaxf(fmaxf(x.x, x.y), fmaxf(x.z, x.w)));
  }
  __shared__ float red[256];
  red[t] = m;
  __syncthreads();
  for (int off = 128; off > 0; off >>= 1) {
    if (t < off) red[t] = fmaxf(red[t], red[t + off]);
    __syncthreads();
  }
  m = red[0];
  __syncthreads();
  float s = 0.f;
#pragma unroll
  for (int i = 0; i < 8; ++i) {
    v4f x = vals[i];
    s += __expf(x.x - m) + __expf(x.y - m) + __expf(x.z - m) + __expf(x.w - m);
  }
  red[t] = s;
  __syncthreads();
  for (int off = 128; off > 0; off >>= 1) {
    if (t < off) red[t] += red[t + off];
    __syncthreads();
  }
  if (t == 0) {
    stats[2 * row]     = m;
    stats[2 * row + 1] = 1.0f / red[0];
  }
}

// ---------------------------------------------------------------------------
// Kernel 2: partial CT[j][i] = sum_{s in split} v[j,s] * ksm[i,s]   (CT = context^T)
// k softmax applied on-the-fly while staging to LDS. WMMA f32 16x16x4.
// Block: 128 threads = 4 waves; wave w owns M-stripe rows 16w..16w+15 of v.
// ---------------------------------------------------------------------------
__global__ void ctx_partial_kernel(const float* __restrict__ k, const float* __restrict__ v,
                                   const float* __restrict__ stats, float* __restrict__ partial) {
  const int head  = blockIdx.x;               // 0..31
  const int split = blockIdx.y;               // 0..7
  const size_t hoff = (size_t)head * D * S;
  const float* kh = k + hoff;
  const float* vh = v + hoff;
  const int t    = threadIdx.x;               // 128
  const int lane = t & 31;
  const int w    = t >> 5;
  const int hf   = lane >> 4;                 // which 16-lane half
  const int lm   = lane & 15;

  __shared__ __align__(16) float vt[D * LDSTR];
  __shared__ __align__(16) float kt[D * LDSTR];
  __shared__ float2 sstat[D];

  if (t < D) {
    sstat[t] = make_float2(stats[2 * (head * D + t)], stats[2 * (head * D + t) + 1]);
  }
  __syncthreads();

  const v8f vzero = {};
  v8f acc[4];
#pragma unroll
  for (int n = 0; n < 4; ++n) acc[n] = vzero;

  const int srow = t >> 4;                    // 0..7 (row within 8-row group)
  const int scol = (t * 4) & 63;              // fixed column (multiple of 4)

  for (int chunk = 0; chunk < SPAN / SCHUNK; ++chunk) {   // 16 chunks of 64 s-values
    const int s0 = split * SPAN + chunk * SCHUNK;
    // Stage 64x64 tiles with b128 loads/stores (coalesced).
#pragma unroll
    for (int i = 0; i < 8; ++i) {
      int row = 8 * i + srow;
      const size_t g = (size_t)row * S + s0 + scol;
      // v: single-use stream -> non-temporal
      v4f vv = __builtin_nontemporal_load((const v4f*)&vh[g]);
      *(v4f*)&vt[row * LDSTR + scol] = vv;
      // k: default policy (second read, should hit L2 after kstats pass)
      v4f kx = *(const v4f*)&kh[g];
      float2 st = sstat[row];
      v4f ks;
      ks.x = __expf(kx.x - st.x) * st.y;
      ks.y = __expf(kx.y - st.x) * st.y;
      ks.z = __expf(kx.z - st.x) * st.y;
      ks.w = __expf(kx.w - st.x) * st.y;
      *(v4f*)&kt[row * LDSTR + scol] = ks;
    }
    __syncthreads();
    // A = v stripe (M=16 rows, K=4), B = ksm (K=4, N=16). Conflict-free b64 LDS reads.
#pragma unroll
    for (int kk = 0; kk < SCHUNK; kk += 4) {
      v2f a = *(const v2f*)&vt[(w * 16 + lm) * LDSTR + kk + 2 * hf];
#pragma unroll
      for (int n = 0; n < 4; ++n) {
        v2f b = *(const v2f*)&kt[(n * 16 + lm) * LDSTR + kk + 2 * hf];
        acc[n] = __builtin_amdgcn_wmma_f32_16x16x4_f32(
            false, a, false, b, (short)0, acc[n], false, false);
      }
    }
    __syncthreads();
  }

  float* p = partial + (size_t)(split * NHEADS + head) * (D * D);
#pragma unroll
  for (int n = 0; n < 4; ++n) {
#pragma unroll
    for (int r = 0; r < 8; ++r) {
      int row = w * 16 + r + 8 * hf;          // M: v row j
      int col = n * 16 + lm;                  // N: k row i
      p[row * D + col] = acc[n][r];
    }
  }
}

// ---------------------------------------------------------------------------
// Kernel 3: deterministic reduction of the 8 split partials -> CT per head
// ---------------------------------------------------------------------------
__global__ void ctx_reduce_kernel(const float* __restrict__ partial, float* __restrict__ ct) {
  int idx = blockIdx.x * blockDim.x + threadIdx.x;
  if (idx >= NHEADS * D * D) return;
  int head = idx / (D * D);
  int off  = idx - head * (D * D);
  float s = 0.f;
#pragma unroll
  for (int sp = 0; sp < SPLITS; ++sp)
    s += partial[(size_t)(sp * NHEADS + head) * (D * D) + off];
  ct[idx] = s;
}

// ---------------------------------------------------------------------------
// Kernel 4: out[j,s] = sum_i CT[j][i] * qsm[i,s]; q softmax (over d) fused in LDS.
// Block: 128 threads = 4 waves handle a 64x128 output tile (wave = 16-row stripe).
// ---------------------------------------------------------------------------
__global__ void out_kernel(const float* __restrict__ q, const float* __restrict__ ct,
                           float* __restrict__ out) {
  const int head  = blockIdx.x;               // 32
  const int stile = blockIdx.y;               // 64 tiles of 128 columns
  const int s0 = stile * 128;
  const float* qh = q + (size_t)head * D * S;
  float* oh = out + (size_t)head * D * S;
  const int t    = threadIdx.x;               // 128
  const int lane = t & 31;
  const int w    = t >> 5;
  const int hf   = lane >> 4;
  const int lm   = lane & 15;

  __shared__ __align__(16) float qs[D * 129]; // stride 129: conflict-free col & row walks
  __shared__ __align__(16) float cts[D * LDSTR];

  // Load CT (64x64) via b128, rows padded to stride 68 (16B aligned).
#pragma unroll
  for (int i = 0; i < 8; ++i) {
    int e   = (i * 128 + t) * 4;              // 0..4092 step 4
    int row = e >> 6;
    int col = e & 63;
    *(v4f*)&cts[row * LDSTR + col] = *(const v4f*)&ct[(size_t)head * (D * D) + e];
  }

  // Fused q softmax over d: thread t owns column s0+t (global reads coalesced, NT).
  float m = -3.4e38f;
  for (int d = 0; d < D; ++d) {
    float x = __builtin_nontemporal_load(&qh[(size_t)d * S + s0 + t]);
    qs[d * 129 + t] = x;
    m = fmaxf(m, x);
  }
  float sum = 0.f;
  for (int d = 0; d < D; ++d) {
    float e = __expf(qs[d * 129 + t] - m);
    sum += e;
    qs[d * 129 + t] = e;
  }
  const float r = 0.125f / sum;               // includes scale d^-1/2 = 1/8
  for (int d = 0; d < D; ++d) qs[d * 129 + t] *= r;
  __syncthreads();

  const int j0 = w * 16;
  for (int st = 0; st < 8; ++st) {            // 8 N-subtiles of 16 columns
    const int sb = st * 16;
    v8f acc = {};
#pragma unroll
    for (int kk = 0; kk < D; kk += 4) {
      v2f a = *(const v2f*)&cts[(j0 + lm) * LDSTR + kk + 2 * hf];
      v2f b;
      b.x = qs[(kk     + 2 * hf) * 129 + sb + lm];
      b.y = qs[(kk + 1 + 2 * hf) * 129 + sb + lm];
      acc = __builtin_amdgcn_wmma_f32_16x16x4_f32(
          false, a, false, b, (short)0, acc, false, false);
    }
#pragma unroll
    for (int rr = 0; rr < 8; ++rr) {
      int row = j0 + rr + 8 * hf;
      __builtin_nontemporal_store(acc[rr], &oh[(size_t)row * S + s0 + sb + lm]);
    }
  }
}

// ---------------------------------------------------------------------------
extern "C" void kernel_launch(void* const* d_in, const int* in_sizes, int n_in,
                              void* d_out, int out_size, void* d_ws, size_t ws_size,
                              hipStream_t stream) {
  (void)in_sizes; (void)n_in; (void)out_size; (void)ws_size;
  const float* q = (const float*)d_in[0];
  const float* k = (const float*)d_in[1];
  const float* v = (const float*)d_in[2];
  float* out = (float*)d_out;
  float* ws  = (float*)d_ws;

  // ws layout (floats): [stats: 2048*2][partial: 8*32*4096][ct: 32*4096] ≈ 4.5 MB
  float* stats   = ws;
  float* partial = ws + 2 * NHEADS * D;
  float* ct      = partial + (size_t)SPLITS * NHEADS * D * D;

  kstats_kernel<<<dim3(NHEADS * D), 256, 0, stream>>>(k, stats);
  ctx_partial_kernel<<<dim3(NHEADS, SPLITS), 128, 0, stream>>>(k, v, stats, partial);
  ctx_reduce_kernel<<<dim3((NHEADS * D * D + 255) / 256), 256, 0, stream>>>(partial, ct);
  out_kernel<<<dim3(NHEADS, S / 128), 128, 0, stream>>>(q, ct, out);
}